// HybridRegression_65481071395642
// MI455X (gfx1250) — compile-verified
//
#include <hip/hip_runtime.h>

// ---------------------------------------------------------------------------
// 15-qubit batched state-vector simulator, fully LDS-resident (CDNA5 320KB WGP
// LDS) with interleaved complex state (ds_*_b64 + packed-f32 gate math), plus
// a WMMA (v_wmma_f32_16x16x4_f32) linear head.
// ---------------------------------------------------------------------------

#define NQ     15
#define NDEPTH 3
#define NBATCH 512
#define DIM    32768            // 2^15 amplitudes
#define TPB    512              // 16 wave32 per block
#define APT    (DIM / TPB)      // 64 amplitudes per thread
#define PPT    (DIM / 2 / TPB)  // 32 gate-pairs per thread

typedef float v2f __attribute__((ext_vector_type(2)));
typedef float v8f __attribute__((ext_vector_type(8)));

// ---------------------------------------------------------------------------
// One block == one batch sample. State lives entirely in LDS as interleaved
// (re, im) pairs: 32768 x 8B = 256KB, + 1KB angle scratch (<320KB WGP LDS).
// ---------------------------------------------------------------------------
__global__ __launch_bounds__(TPB) void sim_kernel(
    const float* __restrict__ state_batch,   // [NBATCH][NQ]
    const float* __restrict__ theta,         // [NDEPTH*NQ]
    float* __restrict__ zq_ws)               // [NBATCH][16]  (k=15 zero pad)
{
    extern __shared__ float smem[];
    v2f*   samp = (v2f*)smem;                // DIM complex amplitudes
    float* sang = (float*)(samp + DIM);      // 256 floats: [0..14]=cenc
                                             // [16..30]=senc [32..76]=cRY
                                             // [80..124]=sRY
    const int b   = blockIdx.x;
    const int tid = threadIdx.x;

    // ---- angle prep (cos/sin of half-angles), once per block --------------
    if (tid < NQ) {
        float h = 0.5f * state_batch[b * NQ + tid];
        sang[tid]      = cosf(h);
        sang[16 + tid] = sinf(h);
    } else if (tid < NQ + NDEPTH * NQ) {
        int idx = tid - NQ;                    // 0..44
        float h = 0.5f * theta[idx];
        sang[32 + idx] = cosf(h);
        sang[80 + idx] = sinf(h);
    }
    __syncthreads();

    // ---- init: product state after RX encoding of |0...0> -----------------
    // amp(i) = (-i)^popcount(i) * prod_q (bit_q(i) ? s_q : c_q)
    float ce[NQ], se[NQ];
#pragma unroll
    for (int q = 0; q < NQ; ++q) { ce[q] = sang[q]; se[q] = sang[16 + q]; }

    // thread owns amplitudes i = tid + 512*j  (bits 0..8 fixed = tid)
    float magfix = 1.0f;
#pragma unroll
    for (int q = 0; q < 9; ++q) magfix *= ((tid >> q) & 1) ? se[q] : ce[q];
    int pcfix = __popc(tid);

    for (int j = 0; j < APT; ++j) {
        int i = tid + (j << 9);
        float m = magfix;
#pragma unroll
        for (int k = 0; k < 6; ++k) m *= ((j >> k) & 1) ? se[9 + k] : ce[9 + k];
        int pc = (pcfix + __popc(j)) & 3;
        v2f val;
        if      (pc == 0) { val.x =  m;  val.y = 0.f; }
        else if (pc == 1) { val.x = 0.f; val.y = -m;  }
        else if (pc == 2) { val.x = -m;  val.y = 0.f; }
        else              { val.x = 0.f; val.y =  m;  }
        samp[i] = val;                        // ds_store_b64, conflict-free
    }
    __syncthreads();

    // ---- 3 layers of 15 shared RY gates; CZ ladder fused as a diagonal ----
    // sign into the q==14 pass; final layer's CZ dropped (|amp|^2 invariant).
    // RY acts identically on re and im -> float2 packed math (v_pk/dual fma).
    for (int d = 0; d < NDEPTH; ++d) {
        for (int q = 0; q < NQ; ++q) {
            float c = sang[32 + d * NQ + q];
            float s = sang[80 + d * NQ + q];
            const bool fuse_cz = (q == NQ - 1) && (d != NDEPTH - 1);
            const int lom = (1 << q) - 1;
            for (int k = 0; k < PPT; ++k) {
                int p  = tid + (k << 9);                       // pair index
                int i0 = ((p >> q) << (q + 1)) | (p & lom);
                int i1 = i0 + (1 << q);
                v2f a0 = samp[i0];                             // ds_load_b64
                v2f a1 = samp[i1];
                v2f n0 = c * a0 - s * a1;
                v2f n1 = s * a0 + c * a1;
                if (fuse_cz) {
                    float g0 = (__popc(i0 & (i0 >> 1) & 0x3FFF) & 1) ? -1.f : 1.f;
                    float g1 = (__popc(i1 & (i1 >> 1) & 0x3FFF) & 1) ? -1.f : 1.f;
                    n0 *= g0; n1 *= g1;
                }
                samp[i0] = n0;                                 // ds_store_b64
                samp[i1] = n1;
            }
            __syncthreads();
        }
    }

    // ---- fused probs + per-qubit Z expectation ----------------------------
    // i = tid + 512*j : bits 0..8 fixed per thread, bits 9..14 vary with j.
    float tot = 0.f;
    float accH[6] = {0.f, 0.f, 0.f, 0.f, 0.f, 0.f};
    for (int j = 0; j < APT; ++j) {
        int i = tid + (j << 9);
        v2f a = samp[i];
        float pr = a.x * a.x + a.y * a.y;
        tot += pr;
#pragma unroll
        for (int k = 0; k < 6; ++k) accH[k] += ((j >> k) & 1) ? -pr : pr;
    }
    float z[NQ];
#pragma unroll
    for (int q = 0; q < 9; ++q) z[q] = ((tid >> q) & 1) ? -tot : tot;
#pragma unroll
    for (int q = 9; q < NQ; ++q) z[q] = accH[q - 9];

    // wave32 reduction, then cross-wave via LDS (state no longer needed)
#pragma unroll
    for (int q = 0; q < NQ; ++q) {
#pragma unroll
        for (int off = 16; off >= 1; off >>= 1)
            z[q] += __shfl_xor(z[q], off, 32);
    }
    __syncthreads();                      // done reading state; reuse smem
    float* sred = smem;                   // 15*16 floats of partials
    int lane = tid & 31, wid = tid >> 5;  // 16 waves
    if (lane == 0) {
#pragma unroll
        for (int q = 0; q < NQ; ++q) sred[q * 16 + wid] = z[q];
    }
    __syncthreads();
    if (tid < NQ) {
        float acc = 0.f;
#pragma unroll
        for (int w = 0; w < 16; ++w) acc += sred[tid * 16 + w];
        zq_ws[b * 16 + tid] = acc;
    }
    if (tid == NQ) zq_ws[b * 16 + NQ] = 0.f;   // K pad for WMMA head
}

// ---------------------------------------------------------------------------
// Linear head via v_wmma_f32_16x16x4_f32:  out = zq[:, ::-1] @ w^T + b
// i.e. out[m] = head_b + sum_k zq[m][k] * head_w[14-k].
// A tile: 16 rows x 4 k (documented 32b A layout: lanes 0-15 M=lane with
// K={0,1}; lanes 16-31 with K={2,3}). B broadcasts w_rev[k] across all 16
// columns, so every column of D equals the desired dot product.
// ---------------------------------------------------------------------------
__global__ __launch_bounds__(1024) void head_kernel(
    const float* __restrict__ zq_ws,     // [NBATCH][16]
    const float* __restrict__ head_w,    // [15]
    const float* __restrict__ head_b,    // [1]
    float* __restrict__ out)             // [NBATCH]
{
    const int tid  = threadIdx.x;
    const int lane = tid & 31;
    const int tile = tid >> 5;                 // 0..31 -> rows tile*16..+15
    const int m    = lane & 15;
    const int row  = tile * 16 + m;
    const int hi   = (lane >> 4) << 1;         // 0 (lanes 0-15) or 2 (16-31)

    v8f acc = {0.f, 0.f, 0.f, 0.f, 0.f, 0.f, 0.f, 0.f};
#pragma unroll
    for (int kstep = 0; kstep < 4; ++kstep) {
        int k = kstep * 4 + hi;
        v2f a, bv;
        a.x  = zq_ws[row * 16 + k];
        a.y  = zq_ws[row * 16 + k + 1];
        bv.x = (k     < NQ) ? head_w[14 - k]       : 0.f;
        bv.y = (k + 1 < NQ) ? head_w[14 - (k + 1)] : 0.f;
        acc = __builtin_amdgcn_wmma_f32_16x16x4_f32(
            /*neg_a=*/false, a, /*neg_b=*/false, bv,
            /*c_mod=*/(short)0, acc, /*reuse_a=*/false, /*reuse_b=*/false);
    }
    float hb = head_b[0];
    // D rows: lane 0 -> M=0..7 in acc[0..7], lane 16 -> M=8..15.
    if ((lane & 15) == 0) {
        int base = tile * 16 + (lane >> 4) * 8;
#pragma unroll
        for (int r = 0; r < 8; ++r) out[base + r] = acc[r] + hb;
    }
}

// ---------------------------------------------------------------------------
extern "C" void kernel_launch(void* const* d_in, const int* in_sizes, int n_in,
                              void* d_out, int out_size, void* d_ws, size_t ws_size,
                              hipStream_t stream) {
    const float* state_batch = (const float*)d_in[0];   // [512][15]
    const float* theta       = (const float*)d_in[1];   // [45]
    const float* head_w      = (const float*)d_in[2];   // [15]
    const float* head_b      = (const float*)d_in[3];   // [1]
    float* out   = (float*)d_out;                       // [512]
    float* zq_ws = (float*)d_ws;                        // [512][16] floats

    // 32768 complex (8B) + 256 scratch floats = 263,168 B (<320KB WGP LDS)
    size_t lds_bytes = (size_t)DIM * sizeof(v2f) + 256 * sizeof(float);
    sim_kernel<<<NBATCH, TPB, lds_bytes, stream>>>(state_batch, theta, zq_ws);
    head_kernel<<<1, 1024, 0, stream>>>(zq_ws, head_w, head_b, out);
}